// CPGM_54022098649919
// MI455X (gfx1250) — compile-verified
//
#include <hip/hip_runtime.h>
#include <math.h>

// ---------------------------------------------------------------------------
// Problem constants (from reference): x (1,3,192,192) f32, C=3
// After AvgPool2d(2,2): 3 x 96 x 96, N = 9216 tokens.
// Dominant op: attention over N=9216 tokens with head dim C=3.
// Strategy: two-pass exact softmax attention; V_WMMA_F32_16X16X4_F32 computes
// S = Q*K^T tiles; NO cross-lane ops inside the streaming loops (row max is
// order-independent -> per-lane partial max in pass 1, per-lane partial
// exp-sum / P*V accumulation in pass 2; one butterfly reduction at the end).
// ---------------------------------------------------------------------------
#define CCH   3
#define H0    192
#define W0    192
#define HH    96
#define WW    96
#define NPIX  (HH * WW)          // 9216
#define KTILES (NPIX / 16)       // 576
#define EPSF  1e-5f

typedef __attribute__((ext_vector_type(2))) float v2f;
typedef __attribute__((ext_vector_type(8))) float v8f;

__device__ __forceinline__ float gelu_exact(float v) {
    return 0.5f * v * (1.0f + erff(v * 0.70710678118654752f));
}
__device__ __forceinline__ float sigmoid_f(float v) {
    return 1.0f / (1.0f + __expf(-v));
}

// ---------------------------------------------------------------------------
// K1: AvgPool2d(2,2):  x[3][192][192] -> xp[3][96][96]
// ---------------------------------------------------------------------------
__global__ void k_pool(const float* __restrict__ x, float* __restrict__ xp) {
    int i = blockIdx.x * blockDim.x + threadIdx.x;
    if (i >= CCH * NPIX) return;
    int w = i % WW, h = (i / WW) % HH, c = i / NPIX;
    const float* xc = x + c * H0 * W0;
    int h2 = h * 2, w2 = w * 2;
    float s = xc[h2 * W0 + w2] + xc[h2 * W0 + w2 + 1] +
              xc[(h2 + 1) * W0 + w2] + xc[(h2 + 1) * W0 + w2 + 1];
    xp[i] = s * 0.25f;
}

// ---------------------------------------------------------------------------
// K2: global average pool per channel: gap[c] = mean(xp[c])
// ---------------------------------------------------------------------------
__global__ void k_gap(const float* __restrict__ xp, float* __restrict__ gap) {
    __shared__ float red[256];
    int c = blockIdx.x;
    float s = 0.f;
    for (int i = threadIdx.x; i < NPIX; i += 256) s += xp[c * NPIX + i];
    red[threadIdx.x] = s;
    __syncthreads();
    for (int off = 128; off > 0; off >>= 1) {
        if (threadIdx.x < off) red[threadIdx.x] += red[threadIdx.x + off];
        __syncthreads();
    }
    if (threadIdx.x == 0) gap[c] = red[0] * (1.0f / (float)NPIX);
}

// ---------------------------------------------------------------------------
// K3: LPG branch -> Kp (prompt_l, padded [N][4]) and Vp (xV2, padded [N][4])
// ---------------------------------------------------------------------------
__global__ void k_lpg(const float* __restrict__ xp, const float* __restrict__ gap,
                      const float* __restrict__ qW, const float* __restrict__ qb,
                      const float* __restrict__ kW, const float* __restrict__ kb,
                      const float* __restrict__ lW, const float* __restrict__ lb,
                      const float* __restrict__ oW, const float* __restrict__ ob,
                      float* __restrict__ Kp, float* __restrict__ Vp) {
    int n = blockIdx.x * blockDim.x + threadIdx.x;
    if (n >= NPIX) return;
    float xt[CCH], xg[CCH], xQ[CCH], xK[CCH], xV[CCH];
#pragma unroll
    for (int c = 0; c < CCH; ++c) {
        xt[c] = xp[c * NPIX + n];
        xg[c] = gap[c] * xt[c];
    }
#pragma unroll
    for (int j = 0; j < CCH; ++j) {
        xQ[j] = qb[j] + xg[0] * qW[j] + xg[1] * qW[3 + j] + xg[2] * qW[6 + j];
        xK[j] = kb[j] + xg[0] * kW[j] + xg[1] * kW[3 + j] + xg[2] * kW[6 + j];
    }
#pragma unroll
    for (int j = 0; j < CCH; ++j) xV[j] = sigmoid_f(xQ[j]) * xK[j] + xg[j];
#pragma unroll
    for (int j = 0; j < CCH; ++j) {
        Kp[n * 4 + j] = lb[j] + xV[0] * lW[j] + xV[1] * lW[3 + j] + xV[2] * lW[6 + j];
        Vp[n * 4 + j] = ob[j] + xt[0] * oW[j] + xt[1] * oW[3 + j] + xt[2] * oW[6 + j];
    }
    Kp[n * 4 + 3] = 0.f;
    Vp[n * 4 + 3] = 0.f;
}

// ---------------------------------------------------------------------------
// K4: HPG branch -> Qp (prompt_h, padded [N][4])
// ---------------------------------------------------------------------------
struct HpgParams {
    const float *sch_dw, *sch_dwb, *sch_pw, *sch_pwb, *sch_bng, *sch_bnb, *sch_bnm, *sch_bnv;
    const float *scv_dw, *scv_dwb, *scv_pw, *scv_pwb, *scv_bng, *scv_bnb, *scv_bnm, *scv_bnv;
    const float *convh_w, *convh_b;
};

__global__ void k_hpg(const float* __restrict__ xp, HpgParams P, float* __restrict__ Qp) {
    int n = blockIdx.x * blockDim.x + threadIdx.x;
    if (n >= NPIX) return;
    int w = n % WW, h = n / WW;
    float ys[CCH], yv[CCH], sum[CCH];
#pragma unroll
    for (int c = 0; c < CCH; ++c) {
        float acc = P.sch_dwb[c];     // (5,1) kernel, pad H by 2 (vertical)
#pragma unroll
        for (int i = 0; i < 5; ++i) {
            int hh = h - 2 + i;
            if (hh >= 0 && hh < HH) acc += xp[c * NPIX + hh * WW + w] * P.sch_dw[c * 5 + i];
        }
        ys[c] = gelu_exact(acc);
        float accv = P.scv_dwb[c];    // (1,5) kernel, pad W by 2 (horizontal)
#pragma unroll
        for (int j = 0; j < 5; ++j) {
            int ww = w - 2 + j;
            if (ww >= 0 && ww < WW) accv += xp[c * NPIX + h * WW + ww] * P.scv_dw[c * 5 + j];
        }
        yv[c] = gelu_exact(accv);
    }
#pragma unroll
    for (int o = 0; o < CCH; ++o) {
        float th = P.sch_pwb[o] + ys[0] * P.sch_pw[o * 3] + ys[1] * P.sch_pw[o * 3 + 1] + ys[2] * P.sch_pw[o * 3 + 2];
        float xh = (th - P.sch_bnm[o]) * P.sch_bng[o] * rsqrtf(P.sch_bnv[o] + EPSF) + P.sch_bnb[o];
        float tv = P.scv_pwb[o] + yv[0] * P.scv_pw[o * 3] + yv[1] * P.scv_pw[o * 3 + 1] + yv[2] * P.scv_pw[o * 3 + 2];
        float xv = (tv - P.scv_bnm[o]) * P.scv_bng[o] * rsqrtf(P.scv_bnv[o] + EPSF) + P.scv_bnb[o];
        sum[o] = xh + xv;
    }
#pragma unroll
    for (int o = 0; o < CCH; ++o)
        Qp[n * 4 + o] = P.convh_b[o] + sum[0] * P.convh_w[o * 3] + sum[1] * P.convh_w[o * 3 + 1]
                                     + sum[2] * P.convh_w[o * 3 + 2];
    Qp[n * 4 + 3] = 0.f;
}

// ---------------------------------------------------------------------------
// K5: two-pass WMMA attention.  One wave per 16-row query tile.
//   S_tile(16x16) = Q_tile(16x4) * K_tile(16x4)^T  via V_WMMA_F32_16X16X4_F32
//   A layout (ISA 7.12.2, 32-bit A 16x4): lanes 0-15 hold K-components {0,1},
//   lanes 16-31 hold {2,3}.  B (4x16) mirrored (B[k][n] = K[n][k]).
//   D layout: VGPR r = row M=r (lanes 0-15, N=lane) / M=r+8 (lanes 16-31).
//   Pass 1: per-lane running max (order-independent; no cross-lane traffic).
//   Pass 2: per-lane partial exp-sum + P*V accumulation.
//   Cross-lane butterfly reductions happen exactly once per pass.
// ---------------------------------------------------------------------------
__global__ __launch_bounds__(32) void k_attn(const float* __restrict__ Qp,
                                             const float* __restrict__ Kp,
                                             const float* __restrict__ Vp,
                                             float* __restrict__ Pout) {
    const int qb   = blockIdx.x << 4;
    const int lane = threadIdx.x & 31;
    const int row  = lane & 15;
    const int hi   = lane >> 4;

    const float scale = 0.57735026918962576f;  // C^(-1/2), folded into Q
    const float2 qv = *(const float2*)(Qp + ((qb + row) << 2) + (hi << 1));
    v2f a;
    a.x = qv.x * scale;
    a.y = qv.y * scale;

    // ---------------- pass 1: exact row max -------------------------------
    float pm[8];
#pragma unroll
    for (int r = 0; r < 8; ++r) pm[r] = -3.0e38f;

    for (int kt = 0; kt < KTILES; ++kt) {
        const int kb = kt << 4;
        const float2 kv = *(const float2*)(Kp + ((kb + row) << 2) + (hi << 1));
        v2f b;
        b.x = kv.x;
        b.y = kv.y;
        v8f cz = {0.f, 0.f, 0.f, 0.f, 0.f, 0.f, 0.f, 0.f};
        v8f s = __builtin_amdgcn_wmma_f32_16x16x4_f32(
            false, a, false, b, (short)0, cz, false, false);
#pragma unroll
        for (int r = 0; r < 8; ++r) pm[r] = fmaxf(pm[r], s[r]);
    }
    // one butterfly max-reduction per row register (stays within 16-lane half)
#pragma unroll
    for (int r = 0; r < 8; ++r) {
        float mx = pm[r];
        mx = fmaxf(mx, __shfl_xor(mx, 1, 32));
        mx = fmaxf(mx, __shfl_xor(mx, 2, 32));
        mx = fmaxf(mx, __shfl_xor(mx, 4, 32));
        mx = fmaxf(mx, __shfl_xor(mx, 8, 32));
        pm[r] = mx;
    }

    // ---------------- pass 2: exp-sum and P*V -----------------------------
    float l[8], ax[8], ay[8], az[8];
#pragma unroll
    for (int r = 0; r < 8; ++r) { l[r] = 0.f; ax[r] = 0.f; ay[r] = 0.f; az[r] = 0.f; }

    for (int kt = 0; kt < KTILES; ++kt) {
        const int kb = kt << 4;
        const float2 kv = *(const float2*)(Kp + ((kb + row) << 2) + (hi << 1));
        v2f b;
        b.x = kv.x;
        b.y = kv.y;
        const float4 vv = *(const float4*)(Vp + ((kb + row) << 2));  // V row of owned column

        v8f cz = {0.f, 0.f, 0.f, 0.f, 0.f, 0.f, 0.f, 0.f};
        v8f s = __builtin_amdgcn_wmma_f32_16x16x4_f32(
            false, a, false, b, (short)0, cz, false, false);

#pragma unroll
        for (int r = 0; r < 8; ++r) {
            float p = __expf(s[r] - pm[r]);
            l[r]  += p;
            ax[r] += p * vv.x;
            ay[r] += p * vv.y;
            az[r] += p * vv.z;
        }
    }

    // final cross-lane sum reductions (once per wave) + write
#pragma unroll
    for (int r = 0; r < 8; ++r) {
        float ls = l[r], sx = ax[r], sy = ay[r], sz = az[r];
#pragma unroll
        for (int msk = 1; msk <= 8; msk <<= 1) {
            ls += __shfl_xor(ls, msk, 32);
            sx += __shfl_xor(sx, msk, 32);
            sy += __shfl_xor(sy, msk, 32);
            sz += __shfl_xor(sz, msk, 32);
        }
        if (row == 0) {
            const float inv = 1.0f / ls;
            const int n = qb + r + (hi << 3);
            float4 o;
            o.x = sx * inv; o.y = sy * inv; o.z = sz * inv; o.w = 0.f;
            *(float4*)(Pout + (n << 2)) = o;
        }
    }
}

// ---------------------------------------------------------------------------
// K6: prompt = lin(P, pW, pb) + xV2 ; LayerNorm over C ; -> outc [3][96][96]
// ---------------------------------------------------------------------------
__global__ void k_post(const float* __restrict__ Pout, const float* __restrict__ Vp,
                       const float* __restrict__ pW, const float* __restrict__ pb,
                       const float* __restrict__ lng, const float* __restrict__ lnb,
                       float* __restrict__ outc) {
    int n = blockIdx.x * blockDim.x + threadIdx.x;
    if (n >= NPIX) return;
    float P0 = Pout[n * 4], P1 = Pout[n * 4 + 1], P2 = Pout[n * 4 + 2];
    float pr[CCH];
#pragma unroll
    for (int j = 0; j < CCH; ++j)
        pr[j] = pb[j] + P0 * pW[j] + P1 * pW[3 + j] + P2 * pW[6 + j] + Vp[n * 4 + j];
    float mu = (pr[0] + pr[1] + pr[2]) * (1.0f / 3.0f);
    float d0 = pr[0] - mu, d1 = pr[1] - mu, d2 = pr[2] - mu;
    float var = (d0 * d0 + d1 * d1 + d2 * d2) * (1.0f / 3.0f);
    float inv = rsqrtf(var + EPSF);
    outc[0 * NPIX + n] = d0 * inv * lng[0] + lnb[0];
    outc[1 * NPIX + n] = d1 * inv * lng[1] + lnb[1];
    outc[2 * NPIX + n] = d2 * inv * lng[2] + lnb[2];
}

// ---------------------------------------------------------------------------
// K7: DSC residual: dw3x3 -> gelu -> pw -> bn, add residual -> fin
// ---------------------------------------------------------------------------
struct DscParams { const float *dw, *dwb, *pw, *pwb, *bng, *bnb, *bnm, *bnv; };

__global__ void k_dsc(const float* __restrict__ outc, DscParams P, float* __restrict__ fin) {
    int n = blockIdx.x * blockDim.x + threadIdx.x;
    if (n >= NPIX) return;
    int w = n % WW, h = n / WW;
    float y[CCH];
#pragma unroll
    for (int c = 0; c < CCH; ++c) {
        float acc = P.dwb[c];
#pragma unroll
        for (int i = 0; i < 3; ++i) {
#pragma unroll
            for (int j = 0; j < 3; ++j) {
                int hh = h - 1 + i, ww = w - 1 + j;
                if (hh >= 0 && hh < HH && ww >= 0 && ww < WW)
                    acc += outc[c * NPIX + hh * WW + ww] * P.dw[c * 9 + i * 3 + j];
            }
        }
        y[c] = gelu_exact(acc);
    }
#pragma unroll
    for (int o = 0; o < CCH; ++o) {
        float t = P.pwb[o] + y[0] * P.pw[o * 3] + y[1] * P.pw[o * 3 + 1] + y[2] * P.pw[o * 3 + 2];
        float z = (t - P.bnm[o]) * P.bng[o] * rsqrtf(P.bnv[o] + EPSF) + P.bnb[o];
        fin[o * NPIX + n] = z + outc[o * NPIX + n];
    }
}

// ---------------------------------------------------------------------------
// K8: bilinear x2 upsample (align_corners=True) -> d_out [3][192][192]
// ---------------------------------------------------------------------------
__global__ void k_up(const float* __restrict__ fin, float* __restrict__ out) {
    int i = blockIdx.x * blockDim.x + threadIdx.x;
    if (i >= CCH * H0 * W0) return;
    int ow = i % W0, oh = (i / W0) % H0, c = i / (H0 * W0);
    const float r = (float)(HH - 1) / (float)(H0 - 1);  // 95/191
    float fh = oh * r, fw = ow * r;
    int h0 = (int)floorf(fh); int h1 = min(h0 + 1, HH - 1); float wh = fh - (float)h0;
    int w0 = (int)floorf(fw); int w1 = min(w0 + 1, WW - 1); float wwf = fw - (float)w0;
    const float* f = fin + c * NPIX;
    float v00 = f[h0 * WW + w0], v01 = f[h0 * WW + w1];
    float v10 = f[h1 * WW + w0], v11 = f[h1 * WW + w1];
    float top = v00 * (1.f - wwf) + v01 * wwf;
    float bot = v10 * (1.f - wwf) + v11 * wwf;
    out[i] = top * (1.f - wh) + bot * wh;
}

// ---------------------------------------------------------------------------
// Host launcher
// ---------------------------------------------------------------------------
extern "C" void kernel_launch(void* const* d_in, const int* in_sizes, int n_in,
                              void* d_out, int out_size, void* d_ws, size_t ws_size,
                              hipStream_t stream) {
    (void)in_sizes; (void)n_in; (void)out_size; (void)ws_size;
    const float* x       = (const float*)d_in[0];
    const float* qW      = (const float*)d_in[1];
    const float* kW      = (const float*)d_in[2];
    const float* lW      = (const float*)d_in[3];
    const float* oW      = (const float*)d_in[4];
    const float* pW      = (const float*)d_in[5];
    const float* qb      = (const float*)d_in[6];
    const float* kb      = (const float*)d_in[7];
    const float* lb      = (const float*)d_in[8];
    const float* ob      = (const float*)d_in[9];
    const float* pb      = (const float*)d_in[10];
    const float* convh_b = (const float*)d_in[11];
    const float* ln_g    = (const float*)d_in[12];
    const float* ln_b    = (const float*)d_in[13];
    const float* convh_w = (const float*)d_in[14];

    HpgParams hp;
    hp.sch_dw  = (const float*)d_in[15]; hp.sch_dwb = (const float*)d_in[16];
    hp.sch_pw  = (const float*)d_in[17]; hp.sch_pwb = (const float*)d_in[18];
    hp.sch_bng = (const float*)d_in[19]; hp.sch_bnb = (const float*)d_in[20];
    hp.sch_bnm = (const float*)d_in[21]; hp.sch_bnv = (const float*)d_in[22];
    hp.scv_dw  = (const float*)d_in[23]; hp.scv_dwb = (const float*)d_in[24];
    hp.scv_pw  = (const float*)d_in[25]; hp.scv_pwb = (const float*)d_in[26];
    hp.scv_bng = (const float*)d_in[27]; hp.scv_bnb = (const float*)d_in[28];
    hp.scv_bnm = (const float*)d_in[29]; hp.scv_bnv = (const float*)d_in[30];
    hp.convh_w = convh_w; hp.convh_b = convh_b;

    DscParams dp;
    dp.dw  = (const float*)d_in[31]; dp.dwb = (const float*)d_in[32];
    dp.pw  = (const float*)d_in[33]; dp.pwb = (const float*)d_in[34];
    dp.bng = (const float*)d_in[35]; dp.bnb = (const float*)d_in[36];
    dp.bnm = (const float*)d_in[37]; dp.bnv = (const float*)d_in[38];

    // workspace layout (floats)
    float* ws = (float*)d_ws;
    float* xp   = ws;                        // 27648
    float* gap  = xp + CCH * NPIX;           // 4 (padded)
    float* Qp   = gap + 4;                   // N*4
    float* Kp   = Qp + NPIX * 4;             // N*4
    float* Vp   = Kp + NPIX * 4;             // N*4
    float* Pout = Vp + NPIX * 4;             // N*4
    float* outc = Pout + NPIX * 4;           // 27648
    float* fin  = outc + CCH * NPIX;         // 27648

    float* out = (float*)d_out;

    k_pool<<<(CCH * NPIX + 255) / 256, 256, 0, stream>>>(x, xp);
    k_gap<<<CCH, 256, 0, stream>>>(xp, gap);
    k_lpg<<<(NPIX + 255) / 256, 256, 0, stream>>>(xp, gap, qW, qb, kW, kb, lW, lb, oW, ob, Kp, Vp);
    k_hpg<<<(NPIX + 255) / 256, 256, 0, stream>>>(xp, hp, Qp);
    k_attn<<<KTILES, 32, 0, stream>>>(Qp, Kp, Vp, Pout);
    k_post<<<(NPIX + 255) / 256, 256, 0, stream>>>(Pout, Vp, pW, pb, ln_g, ln_b, outc);
    k_dsc<<<(NPIX + 255) / 256, 256, 0, stream>>>(outc, dp, fin);
    k_up<<<(CCH * H0 * W0 + 255) / 256, 256, 0, stream>>>(fin, out);
}